// EdgeFusionLayer_58334245814304
// MI455X (gfx1250) — compile-verified
//
#include <hip/hip_runtime.h>
#include <hip/hip_bf16.h>

typedef __attribute__((ext_vector_type(2))) float v2f;
typedef __attribute__((ext_vector_type(8))) float v8f;

// Computes, per edge (row of 8 floats):
//   mean -> e = x + b*mean - c*(x-mean)^2 -> z = e@W^T + bias -> softmax -> dot(alpha, x)
// One wave32 processes a 16-edge tile via V_WMMA_F32_16X16X4_F32 (two K=4 chunks),
// mapped as z^T = W @ e^T so each lane 0..15 ends up owning all 8 logits of one edge.
__global__ __launch_bounds__(256) void edge_fusion_wmma_kernel(
    const float* __restrict__ x,      // [E, 8]
    const float* __restrict__ W,      // [8, 8] row-major (out, in)
    const float* __restrict__ bias,   // [8]
    const float* __restrict__ bscale, // [1]
    const float* __restrict__ cscale, // [1]
    float* __restrict__ out,          // [E]
    int E)
{
    const int lane = threadIdx.x & 31;
    const int wave = threadIdx.x >> 5;
    const int tileBase = (blockIdx.x * 8 + wave) * 16;

    const int n  = lane & 15;         // edge-in-tile (B/D column), also W row for A
    const int kb = (lane >> 4) << 1;  // K base for this half-wave: 0 or 2

    const int row  = tileBase + n;
    const int vrow = (row < E) ? row : (E - 1);   // clamped load row

    // Each lane loads its half-row: elements {kb,kb+1} and {4+kb,5+kb}.
    const float* rp = x + (size_t)vrow * 8u;
    v2f p0 = *(const v2f*)(rp + kb);
    v2f p1 = *(const v2f*)(rp + 4 + kb);

    // Exchange with partner lane (lane ^ 16) so lanes 0..15 can see the full row
    // later; must happen while all lanes are active.
    v2f q0, q1;
    q0.x = __shfl_xor(p0.x, 16, 32);
    q0.y = __shfl_xor(p0.y, 16, 32);
    q1.x = __shfl_xor(p1.x, 16, 32);
    q1.y = __shfl_xor(p1.y, 16, 32);

    // Row mean: each lane has 4 of the 8 elements; partner has the rest.
    float partial = (p0.x + p0.y) + (p1.x + p1.y);
    float mean = (partial + __shfl_xor(partial, 16, 32)) * 0.125f;

    const float bs = bscale[0];
    const float cs = cscale[0];

    // e = x + b*mean - c*(x-mean)^2  -> these become the B-matrix (e^T chunks)
    v2f e0, e1;
    {
        float d;
        d = p0.x - mean; e0.x = p0.x + bs * mean - cs * d * d;
        d = p0.y - mean; e0.y = p0.y + bs * mean - cs * d * d;
        d = p1.x - mean; e1.x = p1.x + bs * mean - cs * d * d;
        d = p1.y - mean; e1.y = p1.y + bs * mean - cs * d * d;
    }

    // A-matrix = W (16x4 chunks, rows 8..15 zero-padded).
    // A layout: VGPR0 = K=kb, VGPR1 = K=kb+1 for M = lane%16.
    const float  msk = (n < 8) ? 1.0f : 0.0f;
    const float* wr  = W + (size_t)(n & 7) * 8u;      // clamped; masked below
    v2f a0 = *(const v2f*)(wr + kb);
    v2f a1 = *(const v2f*)(wr + 4 + kb);
    a0.x *= msk; a0.y *= msk; a1.x *= msk; a1.y *= msk;

    // C = bias[m] broadcast along columns; D lanes>=16 carry the zero-padded
    // rows M=8..15, so their C must be 0.
    v8f acc;
    const float cmask = (lane < 16) ? 1.0f : 0.0f;
#pragma unroll
    for (int i = 0; i < 8; ++i) acc[i] = bias[i] * cmask;

    // z^T = W[:,0:4] @ e^T[0:4,:] + W[:,4:8] @ e^T[4:8,:] + bias
    acc = __builtin_amdgcn_wmma_f32_16x16x4_f32(false, a0, false, e0,
                                                (short)0, acc, false, false);
    acc = __builtin_amdgcn_wmma_f32_16x16x4_f32(false, a1, false, e1,
                                                (short)0, acc, false, false);

    // Epilogue: lanes 0..15 own edge `row`; acc[j] = z_j for j=0..7.
    if (lane < 16 && row < E) {
        // lane<16 has kb==0: p0={x0,x1}, q0={x2,x3}, p1={x4,x5}, q1={x6,x7}
        const float xs[8] = { p0.x, p0.y, q0.x, q0.y, p1.x, p1.y, q1.x, q1.y };
        float zmax = acc[0];
#pragma unroll
        for (int j = 1; j < 8; ++j) zmax = fmaxf(zmax, acc[j]);
        float ssum = 0.0f, dsum = 0.0f;
#pragma unroll
        for (int j = 0; j < 8; ++j) {
            float a = __expf(acc[j] - zmax);
            ssum += a;
            dsum += a * xs[j];
        }
        out[row] = dsum / ssum;
    }
}

extern "C" void kernel_launch(void* const* d_in, const int* in_sizes, int n_in,
                              void* d_out, int out_size, void* d_ws, size_t ws_size,
                              hipStream_t stream)
{
    (void)n_in; (void)d_ws; (void)ws_size;
    const float* edge_attr = (const float*)d_in[0];
    const float* W         = (const float*)d_in[1];
    const float* bias      = (const float*)d_in[2];
    const float* b         = (const float*)d_in[3];
    const float* c         = (const float*)d_in[4];
    float* out = (float*)d_out;

    const int E = out_size;                 // output is [E]
    const int edgesPerBlock = 8 * 16;       // 8 wave32s x 16-edge tiles
    const int blocks = (E + edgesPerBlock - 1) / edgesPerBlock;
    edge_fusion_wmma_kernel<<<blocks, 256, 0, stream>>>(edge_attr, W, bias, b, c, out, E);
}